// MultiSubtableLngramInjector_1717986918579
// MI455X (gfx1250) — compile-verified
//
#include <hip/hip_runtime.h>

// ---- problem constants (from reference) ------------------------------------
#define B_ 2
#define S_ 4
#define T_ 1024
#define H_ 256            // hidden dim == output channels per subtable
#define R_ 32             // H / BITS
#define V_ (T_ - 1)       // T - NG + 1, NG = 2
#define M_ 8
#define ROUTE_VOCAB_ 65536
#define SUB_STRIDE_ (R_ * ROUTE_VOCAB_)   // 2,097,152 rows per subtable

typedef float v2f __attribute__((ext_vector_type(2)));
typedef float v8f __attribute__((ext_vector_type(8)));

// ----------------------------------------------------------------------------
// Kernel 1: logits = hidden @ q_weight[s]^T via fp32 WMMA (16x16x4), then
// sign-pack 8 consecutive o-channels into one byte code per (b,s,t,r).
//
// One wave owns a 16(t) x 16(o) tile and runs K=256 as 64 chained
// v_wmma_f32_16x16x4_f32 ops (D->C accumulation chain, no A/B hazard).
//
// A-matrix (16x4 f32) per-lane layout (ISA 7.12.2):
//   lane L: m = L&15 ; VGPR0/1 = K = 2*(L>>4) + {0,1}
// B-matrix (4x16 f32) mirrors it with n = L&15. C/D (16x16 f32, 8 VGPRs):
//   VGPR j: M = j + 8*(L>>4), N = L&15.
//
// Sign-pack: ballot_w32(acc[j] > 0) bit L corresponds to
//   (t = t0 + j + 8*(L>>4), o = o0 + (L&15))  ==> byte [7:0]   = (t0+j , r0)
//                                                  byte [15:8]  = (t0+j , r0+1)
//                                                  byte [23:16] = (t0+j+8, r0)
//                                                  byte [31:24] = (t0+j+8, r0+1)
// Each of the 32 lanes stores exactly one code byte.
// ----------------------------------------------------------------------------
__global__ __launch_bounds__(256)
void lngram_pack_codes(const float* __restrict__ hidden,
                       const float* __restrict__ qw,
                       unsigned char* __restrict__ codes) {
  const int lane = threadIdx.x & 31;
  const int wave = (blockIdx.x << 3) | (threadIdx.x >> 5);   // 0..8191

  const int ot = wave & 15;          // o-tile  (16 tiles of 16)
  const int tt = (wave >> 4) & 63;   // t-tile  (64 tiles of 16)
  const int s  = (wave >> 10) & 3;
  const int b  = wave >> 12;

  const int t0  = tt << 4;
  const int o0  = ot << 4;
  const int mn  = lane & 15;         // row-in-tile for A, col-in-tile for B
  const int kof = (lane >> 4) << 1;  // K sub-offset per lane half

  const float* __restrict__ arow =
      hidden + (size_t)(b * T_ + t0 + mn) * H_ + kof;
  const float* __restrict__ brow =
      qw + (size_t)(s * H_ + (o0 + mn)) * H_ + kof;

  v8f acc = {};
#pragma unroll 8
  for (int kk = 0; kk < H_; kk += 4) {
    v2f a  = *(const v2f*)(arow + kk);
    v2f bb = *(const v2f*)(brow + kk);
    // 8 args: (neg_a, A, neg_b, B, c_mod, C, reuse_a, reuse_b)
    acc = __builtin_amdgcn_wmma_f32_16x16x4_f32(
        false, a, false, bb, (short)0, acc, false, false);
  }

  // One ballot per accumulator VGPR -> 4 code bytes each.
  unsigned mk0 = __builtin_amdgcn_ballot_w32(acc[0] > 0.0f);
  unsigned mk1 = __builtin_amdgcn_ballot_w32(acc[1] > 0.0f);
  unsigned mk2 = __builtin_amdgcn_ballot_w32(acc[2] > 0.0f);
  unsigned mk3 = __builtin_amdgcn_ballot_w32(acc[3] > 0.0f);
  unsigned mk4 = __builtin_amdgcn_ballot_w32(acc[4] > 0.0f);
  unsigned mk5 = __builtin_amdgcn_ballot_w32(acc[5] > 0.0f);
  unsigned mk6 = __builtin_amdgcn_ballot_w32(acc[6] > 0.0f);
  unsigned mk7 = __builtin_amdgcn_ballot_w32(acc[7] > 0.0f);

  const int jj   = lane & 7;   // which accumulator VGPR
  const int part = lane >> 3;  // which byte of the ballot mask

  unsigned msel = mk0;
  msel = (jj == 1) ? mk1 : msel;
  msel = (jj == 2) ? mk2 : msel;
  msel = (jj == 3) ? mk3 : msel;
  msel = (jj == 4) ? mk4 : msel;
  msel = (jj == 5) ? mk5 : msel;
  msel = (jj == 6) ? mk6 : msel;
  msel = (jj == 7) ? mk7 : msel;

  const unsigned byte = (msel >> (part << 3)) & 0xFFu;
  const int t = t0 + jj + ((part & 2) ? 8 : 0);
  const int r = (ot << 1) + (part & 1);

  codes[(size_t)((b * S_ + s) * T_ + t) * R_ + r] = (unsigned char)byte;
}

// ----------------------------------------------------------------------------
// Kernel 2: bigram address + table gather.
// One thread per (b,s,v,r): addr = code[t=v] | code[t=v+1]<<8,
// row = s*SUB_STRIDE + r*ROUTE_VOCAB + addr, copy 8 floats (2x b128).
// Stores are fully coalesced; loads are the unavoidable random-HBM gather.
// ----------------------------------------------------------------------------
__global__ __launch_bounds__(256)
void lngram_gather(const unsigned char* __restrict__ codes,
                   const float* __restrict__ flat_weight,
                   float* __restrict__ out) {
  const int gid = blockIdx.x * blockDim.x + threadIdx.x;
  if (gid >= B_ * S_ * V_ * R_) return;

  const int r   = gid & (R_ - 1);
  const int tv  = gid >> 5;       // (b*S+s)*V + v
  const int v   = tv % V_;
  const int bs  = tv / V_;        // b*S + s
  const int s   = bs & (S_ - 1);

  const unsigned char* __restrict__ cb =
      codes + (size_t)(bs * T_ + v) * R_ + r;
  const unsigned c0 = cb[0];
  const unsigned c1 = cb[R_];     // next t, same r
  const unsigned addr = c0 | (c1 << 8);

  const size_t row = (size_t)s * SUB_STRIDE_ + (size_t)r * ROUTE_VOCAB_ + addr;
  const float4* __restrict__ src = (const float4*)(flat_weight + row * M_);
  const float4 x = src[0];
  const float4 y = src[1];

  float4* __restrict__ dst = (float4*)(out + (size_t)gid * M_);
  dst[0] = x;
  dst[1] = y;
}

// ----------------------------------------------------------------------------
extern "C" void kernel_launch(void* const* d_in, const int* in_sizes, int n_in,
                              void* d_out, int out_size, void* d_ws, size_t ws_size,
                              hipStream_t stream) {
  const float* hidden      = (const float*)d_in[0];  // (2,1024,256) f32
  const float* q_weight    = (const float*)d_in[1];  // (4,256,256)  f32
  const float* flat_weight = (const float*)d_in[2];  // (8388608,8)  f32
  float* out               = (float*)d_out;          // (2,4,1023,256) f32

  unsigned char* codes = (unsigned char*)d_ws;       // B*S*T*R = 262144 bytes

  // 8192 wave-tiles (b,s,t-tile,o-tile), 8 waves per 256-thread block.
  lngram_pack_codes<<<1024, 256, 0, stream>>>(hidden, q_weight, codes);

  const int total = B_ * S_ * V_ * R_;               // 261,888
  lngram_gather<<<(total + 255) / 256, 256, 0, stream>>>(codes, flat_weight, out);
}